// Upsample_53094385714040
// MI455X (gfx1250) — compile-verified
//
#include <hip/hip_runtime.h>
#include <hip/hip_bf16.h>
#include <stdint.h>

// upfirdn2d upsample x2 with separable kernel [1,3,3,1]/4 per axis.
// Input:  (B*C, 64, 64) f32    Output: (B*C, 128, 128) f32
//
// Memory-bound (335 MB total @ 23.3 TB/s => ~14 us floor). Strategy:
//  - async global->LDS staging of the input tile (CDNA5 TDM-adjacent path:
//    GLOBAL_LOAD_ASYNC_TO_LDS_B128 + s_wait_asynccnt)
//  - separable 2-tap filters in registers (2 FMAs per output)
//  - b128 coalesced output stores (output is 80% of the traffic)

#define IN_W       64
#define IN_H       64
#define OUT_W      128
#define TH         16            // input rows per block
#define LDS_ROWS   (TH + 2)      // staged rows incl. top/bottom halo
#define LDS_STRIDE 72            // floats per LDS row (keeps b128 dsts 16B aligned)
#define COL0       3             // LDS column of input col -1 (left halo)

__global__ __launch_bounds__(256)
void Upsample_53094385714040_kernel(const float* __restrict__ in,
                                    float* __restrict__ out) {
    __shared__ float tile[LDS_ROWS * LDS_STRIDE];

    const int tid = threadIdx.x;
    const int bc  = blockIdx.x;            // channel image index (B*C)
    const int r0  = blockIdx.y * TH;       // first input row of this tile

    const float* __restrict__ src = in  + (size_t)bc * (IN_W * IN_H);
    float* __restrict__       dst = out + (size_t)bc * (OUT_W * OUT_W);

    // ---- zero halo columns (input cols -1 and 64) -------------------------
    if (tid < 2 * LDS_ROWS) {
        const int row = tid >> 1;
        const int col = (tid & 1) ? (COL0 + 1 + IN_W) : COL0;
        tile[row * LDS_STRIDE + col] = 0.0f;
    }

    // ---- stage rows r0-1 .. r0+TH (18 x 64 f32) ---------------------------
    // 18 rows * 16 float4 = 288 b128 transfers, 256 threads -> <=2 per thread.
    const unsigned lds_base = (unsigned)(size_t)(&tile[0]);   // LDS byte offset
    const uint64_t gbase    = (uint64_t)(uintptr_t)src;

#pragma unroll
    for (int it = 0; it < 2; ++it) {
        const int idx = tid + it * 256;
        if (idx < LDS_ROWS * 16) {
            const int row  = idx >> 4;          // 0..17  (LDS row)
            const int col4 = idx & 15;          // float4 column
            const int g    = r0 - 1 + row;      // global input row
            const unsigned lds_off =
                lds_base + (unsigned)((row * LDS_STRIDE + COL0 + 1 + col4 * 4) * 4);
            if ((unsigned)g < (unsigned)IN_H) {
                const unsigned goff = (unsigned)((g * IN_W + col4 * 4) * 4);
                asm volatile(
                    "global_load_async_to_lds_b128 %0, %1, %2"
                    :
                    : "v"(lds_off), "v"(goff), "s"(gbase)
                    : "memory");
            } else {
                // out-of-image row -> zeros (b128 LDS store, 16B aligned)
                float4* p = (float4*)((char*)tile +
                    (size_t)((row * LDS_STRIDE + COL0 + 1 + col4 * 4) * 4));
                *p = make_float4(0.f, 0.f, 0.f, 0.f);
            }
        }
    }

    asm volatile("s_wait_asynccnt 0x0" ::: "memory");
    __syncthreads();

    // ---- compute: each thread -> 4 input px -> 2x8 output strip -----------
    const int lr  = tid >> 4;              // input row within tile (0..15)
    const int lc4 = (tid & 15) << 2;       // first input col (0..60)

    // LDS row lr   = input row (r0+lr-1)  (top neighbor)
    // LDS row lr+1 = input row (r0+lr)
    // LDS row lr+2 = input row (r0+lr+1)  (bottom neighbor)
    const float* rowA = &tile[(lr    ) * LDS_STRIDE + COL0 + lc4];
    const float* rowB = &tile[(lr + 1) * LDS_STRIDE + COL0 + lc4];
    const float* rowD = &tile[(lr + 2) * LDS_STRIDE + COL0 + lc4];

    // vertical 2-tap pass over 6 columns (input cols lc4-1 .. lc4+4)
    float t[6], u[6];
#pragma unroll
    for (int k = 0; k < 6; ++k) {
        const float a = rowA[k];
        const float b = rowB[k];
        const float d = rowD[k];
        t[k] = fmaf(0.25f, a, 0.75f * b);   // even output row
        u[k] = fmaf(0.25f, d, 0.75f * b);   // odd  output row
    }

    // horizontal 2-tap pass -> 8 output columns per row
    float ot[8], ob[8];
#pragma unroll
    for (int j = 0; j < 4; ++j) {
        ot[2 * j]     = fmaf(0.25f, t[j],     0.75f * t[j + 1]);
        ot[2 * j + 1] = fmaf(0.25f, t[j + 2], 0.75f * t[j + 1]);
        ob[2 * j]     = fmaf(0.25f, u[j],     0.75f * u[j + 1]);
        ob[2 * j + 1] = fmaf(0.25f, u[j + 2], 0.75f * u[j + 1]);
    }

    const int oy = (r0 + lr) * 2;
    const int ox = lc4 * 2;
    float4* o0 = (float4*)(dst + (size_t)oy * OUT_W + ox);
    float4* o1 = (float4*)(dst + (size_t)(oy + 1) * OUT_W + ox);
    o0[0] = make_float4(ot[0], ot[1], ot[2], ot[3]);
    o0[1] = make_float4(ot[4], ot[5], ot[6], ot[7]);
    o1[0] = make_float4(ob[0], ob[1], ob[2], ob[3]);
    o1[1] = make_float4(ob[4], ob[5], ob[6], ob[7]);
}

extern "C" void kernel_launch(void* const* d_in, const int* in_sizes, int n_in,
                              void* d_out, int out_size, void* d_ws, size_t ws_size,
                              hipStream_t stream) {
    (void)n_in; (void)out_size; (void)d_ws; (void)ws_size;
    const float* in  = (const float*)d_in[0];   // (8,512,64,64) f32
    float*       out = (float*)d_out;           // (8,512,128,128) f32

    const int nchan = in_sizes[0] / (IN_W * IN_H);   // 8*512 = 4096
    dim3 grid(nchan, IN_H / TH, 1);                  // (4096, 4)
    dim3 block(256, 1, 1);                           // 8 waves (wave32)
    hipLaunchKernelGGL(Upsample_53094385714040_kernel, grid, block, 0, stream,
                       in, out);
}